// RobustConv_56719338111765
// MI455X (gfx1250) — compile-verified
//
#include <hip/hip_runtime.h>
#include <hip/hip_bf16.h>

// RobustConv (GraphConv with variance-attention) for MI455X / gfx1250.
// Pipeline: degree atomics -> fp32 WMMA GEMM (+fused epilogue) -> edge scatter
// (native global_atomic_add_f32 via inline asm) -> final degree scaling.

typedef float v2f __attribute__((ext_vector_type(2)));
typedef float v8f __attribute__((ext_vector_type(8)));

#define N_NODES 50000
#define N_EDGES 800000
#define IN_F    256
#define OUT_F   64
#define N_TILES (N_NODES / 16)   // 3125 exactly
// K-pair-interleaved LDS layout: pair stride 160 floats (== 32 mod 64) makes
// the hi half-wave (pair p+1) use banks 32..63 while the lo half uses 0..31.
#define PAIR_STRIDE 160

// ---------------------------------------------------------------- degrees ----
__global__ __launch_bounds__(256) void rgc_deg_init(unsigned* __restrict__ dgo,
                                                    unsigned* __restrict__ dgi) {
    int i = blockIdx.x * 256 + threadIdx.x;
    if (i < N_NODES) { dgo[i] = 1u; dgi[i] = 1u; }  // self-loop counts
}

__global__ __launch_bounds__(256) void rgc_deg_edges(const int* __restrict__ esrc,
                                                     const int* __restrict__ edst,
                                                     unsigned* __restrict__ dgo,
                                                     unsigned* __restrict__ dgi) {
    int i = blockIdx.x * 256 + threadIdx.x;
    if (i < N_EDGES) {
        atomicAdd(&dgo[esrc[i]], 1u);
        atomicAdd(&dgi[edst[i]], 1u);
    }
}

// ------------------------------------------------------------------- GEMM ----
// One wave computes a 16-row x 64-col tile for BOTH weight matrices using
// V_WMMA_F32_16X16X4_F32. Weights staged in LDS in 64-K-row chunks with
// K-pair-interleaved layout so a single b64 LDS load forms each B fragment.
__global__ __launch_bounds__(256) void rgc_gemm(
    const float*    __restrict__ feat,
    const float*    __restrict__ Wm,
    const float*    __restrict__ Wv,
    const unsigned* __restrict__ dgo,
    float* __restrict__ Am,  float* __restrict__ Av,
    float* __restrict__ outm, float* __restrict__ outv)
{
    __shared__ float sW[2][32 * PAIR_STRIDE];   // 40 KB

    const int lane = threadIdx.x & 31;
    const int wave = threadIdx.x >> 5;
    const int hi   = lane >> 4;       // half-wave id (0/1)
    const int lo   = lane & 15;
    const int tile = blockIdx.x * 8 + wave;
    const bool active = (tile < N_TILES);
    const int m0 = active ? tile * 16 : 0;   // clamp so loads stay in bounds

    v8f accm[4], accv[4];
    const v8f vzero = {0.f, 0.f, 0.f, 0.f, 0.f, 0.f, 0.f, 0.f};
#pragma unroll
    for (int t = 0; t < 4; ++t) { accm[t] = vzero; accv[t] = vzero; }

    // A fragment source: row = m0+lo, K pair starts at 2*hi
    const float* arow = feat + (size_t)(m0 + lo) * IN_F + 2 * hi;

    for (int c = 0; c < 4; ++c) {            // 4 chunks of 64 K-rows
        __syncthreads();                     // previous chunk fully consumed
        // stage 64 rows of both weights: element (k,col) -> pair-interleaved
        for (int i = threadIdx.x; i < (64 * OUT_F) / 4; i += 256) {
            const int k   = (4 * i) >> 6;    // 0..63
            const int col = (4 * i) & 63;
            const float4 m = *(const float4*)(Wm + (size_t)(c * 64 + k) * OUT_F + col);
            const float4 v = *(const float4*)(Wv + (size_t)(c * 64 + k) * OUT_F + col);
            const int base = (k >> 1) * PAIR_STRIDE + col * 2 + (k & 1);
            sW[0][base + 0] = m.x; sW[0][base + 2] = m.y;
            sW[0][base + 4] = m.z; sW[0][base + 6] = m.w;
            sW[1][base + 0] = v.x; sW[1][base + 2] = v.y;
            sW[1][base + 4] = v.z; sW[1][base + 6] = v.w;
        }
        __syncthreads();

#pragma unroll 2
        for (int k = 0; k < 64; k += 4) {
            const v2f a = *(const v2f*)(arow + c * 64 + k);
            // B fragment: rows (k+2*hi, k+2*hi+1) at col -> one contiguous b64
            const int pb = ((k >> 1) + hi) * PAIR_STRIDE + 2 * lo;
#pragma unroll
            for (int t = 0; t < 4; ++t) {
                const v2f bm = *(const v2f*)(&sW[0][pb + 32 * t]);
                const v2f bv = *(const v2f*)(&sW[1][pb + 32 * t]);
                accm[t] = __builtin_amdgcn_wmma_f32_16x16x4_f32(
                    false, a, false, bm, (short)0, accm[t], false, false);
                accv[t] = __builtin_amdgcn_wmma_f32_16x16x4_f32(
                    false, a, false, bv, (short)0, accv[t], false, false);
            }
        }
    }

    if (active) {
        // Fused epilogue: relu, attention=exp(-var), pre-apply src-degree norm,
        // write gather tables and initialize output with self-loop contribution.
#pragma unroll
        for (int v = 0; v < 8; ++v) {
            const int r = m0 + v + 8 * hi;          // C/D layout: M = v + 8*hi
            const float dg   = (float)dgo[r];
            const float rso  = rsqrtf(dg);
            const float invo = 1.0f / dg;
#pragma unroll
            for (int t = 0; t < 4; ++t) {
                const float me  = fmaxf(accm[t][v], 0.0f);
                const float va  = fmaxf(accv[t][v], 0.0f);
                const float att = __expf(-va);
                const float am  = me * att * rso;          // mean*att*rsqrt(deg_out)
                const float av  = va * att * att * invo;   // var*att^2 / deg_out
                const size_t off = (size_t)r * OUT_F + t * 16 + lo;
                Am[off]   = am;
                Av[off]   = av;
                outm[off] = am;   // self-loop edge (s=d=r) contribution
                outv[off] = av;
            }
        }
    }
}

// ---------------------------------------------------------------- scatter ----
// One wave per edge: 32 lanes x float2 cover 64 features of both tensors.
// Native no-return f32 atomics (fire-and-forget, drained at s_endpgm).
__device__ __forceinline__ void gatomic_add2(float* p, float x, float y) {
    asm volatile("global_atomic_add_f32 %0, %1, off"          :: "v"(p), "v"(x) : "memory");
    asm volatile("global_atomic_add_f32 %0, %1, off offset:4" :: "v"(p), "v"(y) : "memory");
}

__global__ __launch_bounds__(256) void rgc_scatter(
    const int*   __restrict__ esrc, const int* __restrict__ edst,
    const float* __restrict__ Am,   const float* __restrict__ Av,
    float* __restrict__ outm, float* __restrict__ outv)
{
    const int e = blockIdx.x * 8 + (threadIdx.x >> 5);
    if (e >= N_EDGES) return;
    const int lane = threadIdx.x & 31;
    const int s = esrc[e];
    const int d = edst[e];
    const float2 m = *(const float2*)(Am + (size_t)s * OUT_F + 2 * lane);
    const float2 v = *(const float2*)(Av + (size_t)s * OUT_F + 2 * lane);
    gatomic_add2(outm + (size_t)d * OUT_F + 2 * lane, m.x, m.y);
    gatomic_add2(outv + (size_t)d * OUT_F + 2 * lane, v.x, v.y);
}

// --------------------------------------------------------------- finalize ----
__global__ __launch_bounds__(256) void rgc_finalize(const unsigned* __restrict__ dgi,
                                                    float* __restrict__ outm,
                                                    float* __restrict__ outv) {
    const long long i = (long long)blockIdx.x * 256 + threadIdx.x;
    if (i < (long long)N_NODES * OUT_F) {
        const float di = (float)dgi[i >> 6];
        outm[i] *= rsqrtf(di);      // * rsqrt(deg_in[d])
        outv[i] *= 1.0f / di;       // * 1/deg_in[d]
    }
}

// ----------------------------------------------------------------- launch ----
extern "C" void kernel_launch(void* const* d_in, const int* in_sizes, int n_in,
                              void* d_out, int out_size, void* d_ws, size_t ws_size,
                              hipStream_t stream) {
    const float* feat = (const float*)d_in[0];
    const float* Wm   = (const float*)d_in[1];
    const float* Wv   = (const float*)d_in[2];
    const int*   esrc = (const int*)d_in[3];
    const int*   edst = (const int*)d_in[4];

    float* outm = (float*)d_out;                         // [N, 64]
    float* outv = outm + (size_t)N_NODES * OUT_F;        // [N, 64]

    // Workspace layout: Am[N*64] f32 | Av[N*64] f32 | dgo[N] u32 | dgi[N] u32
    float*    Am  = (float*)d_ws;
    float*    Av  = Am + (size_t)N_NODES * OUT_F;
    unsigned* dgo = (unsigned*)(Av + (size_t)N_NODES * OUT_F);
    unsigned* dgi = dgo + N_NODES;

    rgc_deg_init <<<(N_NODES + 255) / 256, 256, 0, stream>>>(dgo, dgi);
    rgc_deg_edges<<<(N_EDGES + 255) / 256, 256, 0, stream>>>(esrc, edst, dgo, dgi);
    rgc_gemm     <<<(N_TILES + 7) / 8,     256, 0, stream>>>(feat, Wm, Wv, dgo,
                                                             Am, Av, outm, outv);
    rgc_scatter  <<<N_EDGES / 8,           256, 0, stream>>>(esrc, edst, Am, Av,
                                                             outm, outv);
    rgc_finalize <<<((N_NODES * OUT_F) + 255) / 256, 256, 0, stream>>>(dgi, outm, outv);
}